// Network_61658550501610
// MI455X (gfx1250) — compile-verified
//
#include <hip/hip_runtime.h>
#include <hip/hip_bf16.h>
#include <math.h>

// ---------------- model dims ----------------
#define DM   41    // D_MODEL
#define DI   82    // D_INNER
#define D2I  164   // 2*D_INNER
#define DTR  3     // DT_RANK
#define DST  16    // D_STATE
#define XDBL 35    // DTR + 2*DST
#define HID  64
#define NLB  10
#define BB   16
#define LL   2048
#define NT   (BB*LL)   // 32768 tokens

typedef __attribute__((ext_vector_type(2))) float v2f;
typedef __attribute__((ext_vector_type(8))) float v8f;

#define WMMA_F32X4(c, a, b) \
  (c) = __builtin_amdgcn_wmma_f32_16x16x4_f32(false, (a), false, (b), (short)0, (c), false, false)

__device__ __forceinline__ float sigmoidf_(float v) { return 1.f / (1.f + __expf(-v)); }

// ============================================================
// K1: xz = x @ W_in^T  (NT x 41) * (41 x 164)
//     writes xi = xz[:, :82], s = silu(xz[:, 82:164])
// One wave per 16-token tile, fp32 WMMA 16x16x4, K padded 41->44, N 164->176.
// ============================================================
__global__ __launch_bounds__(256) void k_in_proj(
    const float* __restrict__ x, const float* __restrict__ W_in,
    float* __restrict__ xi, float* __restrict__ s) {
  __shared__ float w[D2I * 44];                    // W_in padded [164][44]
  for (int i = threadIdx.x; i < D2I * 44; i += blockDim.x) {
    int n = i / 44, k = i % 44;
    w[i] = (k < DM) ? W_in[n * DM + k] : 0.f;
  }
  __syncthreads();

  int wave = threadIdx.x >> 5;
  int lane = threadIdx.x & 31;
  int m    = lane & 15;
  int kh   = (lane >> 4) << 1;                     // 0 or 2 (K-half)
  int tile = blockIdx.x * 8 + wave;                // 2048 tiles total
  int t0   = tile * 16;

  const float* arow = x + (size_t)(t0 + m) * DM;
  v2f a[11];
  for (int ks = 0; ks < 11; ++ks) {
    int k = ks * 4 + kh;
    a[ks][0] = (k     < DM) ? arow[k]     : 0.f;
    a[ks][1] = (k + 1 < DM) ? arow[k + 1] : 0.f;
  }

  for (int nt = 0; nt < 11; ++nt) {                // N = 176 padded
    int n = nt * 16 + m;                           // B-fragment column
    bool nv = n < D2I;
    v8f c = {};
    for (int ks = 0; ks < 11; ++ks) {
      v2f b;
      b[0] = nv ? w[n * 44 + ks * 4 + kh]     : 0.f;
      b[1] = nv ? w[n * 44 + ks * 4 + kh + 1] : 0.f;
      WMMA_F32X4(c, a[ks], b);
    }
    int col = nt * 16 + m;                         // D-fragment column == m
    if (col < D2I) {
      int rbase = (lane >> 4) << 3;
      for (int r = 0; r < 8; ++r) {
        int tok = t0 + r + rbase;
        float v = c[r];
        if (col < DI) xi[(size_t)tok * DI + col] = v;
        else          s [(size_t)tok * DI + (col - DI)] = v * sigmoidf_(v);
      }
    }
  }
}

// ============================================================
// K2: depthwise conv(4) + silu -> xc ; x_dbl = xc @ W_xproj^T ;
//     delta = softplus(dt @ W_dt^T + b_dt) ; split Bs, Cs.
// 64 tokens per block (divides L, never crosses batch rows), staged in LDS.
// ============================================================
#define TOK 64
__global__ __launch_bounds__(256) void k_conv_xdbl(
    const float* __restrict__ xi, const float* __restrict__ conv_w,
    const float* __restrict__ conv_b, const float* __restrict__ W_xproj,
    const float* __restrict__ W_dt, const float* __restrict__ b_dt,
    float* __restrict__ xcg, float* __restrict__ delta,
    float* __restrict__ Bsv, float* __restrict__ Csv) {
  __shared__ float xc[TOK][DI];
  __shared__ float xd[TOK][XDBL];
  int t0 = blockIdx.x * TOK;
  int l0 = t0 & (LL - 1);

  for (int i = threadIdx.x; i < TOK * DI; i += blockDim.x) {
    int tt = i / DI, d = i % DI;
    int l = l0 + tt;
    float acc = conv_b[d];
    for (int k = 0; k < 4; ++k) {
      int lp = l - 3 + k;
      float v = (lp >= 0) ? xi[(size_t)(t0 + tt - 3 + k) * DI + d] : 0.f;
      acc += v * conv_w[d * 4 + k];
    }
    float v = acc * sigmoidf_(acc);                // silu
    xc[tt][d] = v;
    xcg[(size_t)(t0 + tt) * DI + d] = v;
  }
  __syncthreads();

  for (int i = threadIdx.x; i < TOK * XDBL; i += blockDim.x) {
    int tt = i / XDBL, e = i % XDBL;
    const float* wr = W_xproj + e * DI;
    float acc = 0.f;
    for (int d = 0; d < DI; ++d) acc += xc[tt][d] * wr[d];
    xd[tt][e] = acc;
    if (e >= DTR && e < DTR + DST)  Bsv[(size_t)(t0 + tt) * DST + (e - DTR)] = acc;
    else if (e >= DTR + DST)        Csv[(size_t)(t0 + tt) * DST + (e - DTR - DST)] = acc;
  }
  __syncthreads();

  for (int i = threadIdx.x; i < TOK * DI; i += blockDim.x) {
    int tt = i / DI, d = i % DI;
    float acc = b_dt[d];
    for (int r = 0; r < DTR; ++r) acc += xd[tt][r] * W_dt[d * DTR + r];
    float sp = (acc > 20.f) ? acc : __logf(1.f + __expf(acc));   // softplus
    delta[(size_t)(t0 + tt) * DI + d] = sp;
  }
}

// ============================================================
// K3: selective scan. One lane per (b, d, n): h in a register over L=2048.
// Wave32 = 2 channels x 16 states; reduce over n via __shfl_xor width 16.
// 656 waves total; all operands are L2-resident.
// ============================================================
__global__ __launch_bounds__(256) void k_scan(
    const float* __restrict__ delta, const float* __restrict__ xcg,
    const float* __restrict__ Bsv, const float* __restrict__ Csv,
    const float* __restrict__ s, const float* __restrict__ A_log,
    const float* __restrict__ Dp, float* __restrict__ y) {
  int tid  = blockIdx.x * blockDim.x + threadIdx.x;
  int wave = tid >> 5;                 // 0..655
  int lane = tid & 31;
  int n    = lane & 15;
  int half = lane >> 4;
  int b    = wave / 41;
  int d    = (wave % 41) * 2 + half;   // 0..81

  float An = -__expf(A_log[d * DST + n]);
  float Dd = Dp[d];
  float h  = 0.f;
  size_t base = (size_t)b * LL;
  for (int l = 0; l < LL; ++l) {
    size_t t = base + l;
    float dl  = delta[t * DI + d];
    float xcv = xcg [t * DI + d];
    float bs  = Bsv [t * DST + n];
    float cs  = Csv [t * DST + n];
    h = __expf(dl * An) * h + dl * xcv * bs;
    float yv = h * cs;
    yv += __shfl_xor(yv, 1, 16);
    yv += __shfl_xor(yv, 2, 16);
    yv += __shfl_xor(yv, 4, 16);
    yv += __shfl_xor(yv, 8, 16);
    if (n == 0)
      y[t * DI + d] = (yv + xcv * Dd) * s[t * DI + d];
  }
}

// ============================================================
// K4: fused head: out = y@W_out^T ; h = relu(out@W_c1^T+b1) ; logits = h@W_c2^T+b2
// One wave per 16-token tile; three chained fp32 WMMA GEMMs through an LDS
// transpose stage (C-layout -> A-layout). Weights zero-padded in LDS.
// ============================================================
__global__ __launch_bounds__(128) void k_head(
    const float* __restrict__ y, const float* __restrict__ W_out,
    const float* __restrict__ W_c1, const float* __restrict__ b_c1,
    const float* __restrict__ W_c2, const float* __restrict__ b_c2,
    float* __restrict__ out) {
  __shared__ float wout[48 * 84];      // [48][84], rows>=41 / cols>=82 zero
  __shared__ float wc1 [64 * 44];      // [64][44], cols>=41 zero
  __shared__ float wc2 [16 * 64];      // [16][64], rows>=10 zero
  __shared__ float bc1 [64];
  __shared__ float bc2 [16];
  __shared__ float stage[4][16 * 65];  // per-wave 16x64 (stride 65) transpose stage

  for (int i = threadIdx.x; i < 48 * 84; i += blockDim.x) {
    int nn = i / 84, k = i % 84;
    wout[i] = (nn < DM && k < DI) ? W_out[nn * DI + k] : 0.f;
  }
  for (int i = threadIdx.x; i < 64 * 44; i += blockDim.x) {
    int nn = i / 44, k = i % 44;
    wc1[i] = (k < DM) ? W_c1[nn * DM + k] : 0.f;
  }
  for (int i = threadIdx.x; i < 16 * 64; i += blockDim.x) {
    int nn = i / 64, k = i % 64;
    wc2[i] = (nn < NLB) ? W_c2[nn * 64 + k] : 0.f;
  }
  for (int i = threadIdx.x; i < 64; i += blockDim.x) bc1[i] = b_c1[i];
  for (int i = threadIdx.x; i < 16; i += blockDim.x) bc2[i] = (i < NLB) ? b_c2[i] : 0.f;
  __syncthreads();

  int wv   = threadIdx.x >> 5;
  int lane = threadIdx.x & 31;
  int m    = lane & 15;
  int kh   = (lane >> 4) << 1;
  int rbase = (lane >> 4) << 3;
  int tile = blockIdx.x * 4 + wv;
  int t0   = tile * 16;
  float* stg = stage[wv];

  // ---- G1: out16x48 = y16x82(pad84) @ wout^T ----
  {
    const float* yrow = y + (size_t)(t0 + m) * DI;
    v2f a[21];
    for (int ks = 0; ks < 21; ++ks) {
      int k = ks * 4 + kh;
      a[ks][0] = (k     < DI) ? yrow[k]     : 0.f;
      a[ks][1] = (k + 1 < DI) ? yrow[k + 1] : 0.f;
    }
    for (int nt = 0; nt < 3; ++nt) {
      int n = nt * 16 + m;
      v8f c = {};
      for (int ks = 0; ks < 21; ++ks) {
        v2f b;
        b[0] = wout[n * 84 + ks * 4 + kh];
        b[1] = wout[n * 84 + ks * 4 + kh + 1];
        WMMA_F32X4(c, a[ks], b);
      }
      for (int r = 0; r < 8; ++r) stg[(r + rbase) * 65 + nt * 16 + m] = c[r];
    }
  }
  __syncthreads();

  // ---- G2: h16x64 = relu(out16x41(pad44) @ wc1^T + b1) ----
  {
    v2f a[11];
    for (int ks = 0; ks < 11; ++ks) {
      int k = ks * 4 + kh;
      a[ks][0] = stg[m * 65 + k];
      a[ks][1] = stg[m * 65 + k + 1];
    }
    __syncthreads();                                 // reads done before overwrite
    for (int nt = 0; nt < 4; ++nt) {
      int n = nt * 16 + m;
      v8f c = {};
      for (int ks = 0; ks < 11; ++ks) {
        v2f b;
        b[0] = wc1[n * 44 + ks * 4 + kh];
        b[1] = wc1[n * 44 + ks * 4 + kh + 1];
        WMMA_F32X4(c, a[ks], b);
      }
      float bias = bc1[nt * 16 + m];
      for (int r = 0; r < 8; ++r) {
        float v = c[r] + bias;
        stg[(r + rbase) * 65 + nt * 16 + m] = v > 0.f ? v : 0.f;
      }
    }
  }
  __syncthreads();

  // ---- G3: logits16x10 = h16x64 @ wc2^T + b2 ----
  {
    v2f a[16];
    for (int ks = 0; ks < 16; ++ks) {
      int k = ks * 4 + kh;
      a[ks][0] = stg[m * 65 + k];
      a[ks][1] = stg[m * 65 + k + 1];
    }
    v8f c = {};
    for (int ks = 0; ks < 16; ++ks) {
      v2f b;
      b[0] = wc2[m * 64 + ks * 4 + kh];
      b[1] = wc2[m * 64 + ks * 4 + kh + 1];
      WMMA_F32X4(c, a[ks], b);
    }
    if (m < NLB) {
      float bias = bc2[m];
      for (int r = 0; r < 8; ++r)
        out[(size_t)(t0 + r + rbase) * NLB + m] = c[r] + bias;
    }
  }
}

// ============================================================
extern "C" void kernel_launch(void* const* d_in, const int* in_sizes, int n_in,
                              void* d_out, int out_size, void* d_ws, size_t ws_size,
                              hipStream_t stream) {
  (void)in_sizes; (void)n_in; (void)out_size; (void)ws_size;
  const float* x       = (const float*)d_in[0];
  const float* W_in    = (const float*)d_in[1];
  const float* conv_w  = (const float*)d_in[2];
  const float* conv_b  = (const float*)d_in[3];
  const float* W_xproj = (const float*)d_in[4];
  const float* W_dt    = (const float*)d_in[5];
  const float* b_dt    = (const float*)d_in[6];
  const float* A_log   = (const float*)d_in[7];
  const float* Dp      = (const float*)d_in[8];
  const float* W_out   = (const float*)d_in[9];
  const float* W_c1    = (const float*)d_in[10];
  const float* b_c1    = (const float*)d_in[11];
  const float* W_c2    = (const float*)d_in[12];
  const float* b_c2    = (const float*)d_in[13];
  float* out = (float*)d_out;

  float* ws    = (float*)d_ws;
  float* xi    = ws;                       // NT*82
  float* s     = xi    + (size_t)NT * DI;  // NT*82
  float* xc    = s     + (size_t)NT * DI;  // NT*82
  float* delta = xc    + (size_t)NT * DI;  // NT*82
  float* Bsv   = delta + (size_t)NT * DI;  // NT*16
  float* Csv   = Bsv   + (size_t)NT * DST; // NT*16
  float* y     = Csv   + (size_t)NT * DST; // NT*82

  k_in_proj  <<<NT / 16 / 8, 256, 0, stream>>>(x, W_in, xi, s);
  k_conv_xdbl<<<NT / TOK,    256, 0, stream>>>(xi, conv_w, conv_b, W_xproj, W_dt, b_dt,
                                               xc, delta, Bsv, Csv);
  k_scan     <<<82,          256, 0, stream>>>(delta, xc, Bsv, Csv, s, A_log, Dp, y);
  k_head     <<<NT / 16 / 4, 128, 0, stream>>>(y, W_out, W_c1, b_c1, W_c2, b_c2, out);
}